// C4MoEExecutor_62380105007556
// MI455X (gfx1250) — compile-verified
//
#include <hip/hip_runtime.h>
#include <stdint.h>

// -------- CDNA5 async global<->LDS path detection (device pass only) --------
#ifndef __HIP_DEVICE_COMPILE__
#define DEV_ASYNC 0
#else
#if __has_builtin(__builtin_amdgcn_global_load_async_to_lds_b128) && \
    __has_builtin(__builtin_amdgcn_global_store_async_from_lds_b128)
#define DEV_ASYNC 1
#else
#define DEV_ASYNC 0
#endif
#endif

#define GAS __attribute__((address_space(1)))
#define LAS __attribute__((address_space(3)))

typedef int v4i __attribute__((__vector_size__(4 * sizeof(int))));

static constexpr int MEMN = 512;   // words per VM memory row
static constexpr int RPB  = 8;     // rows per block, one wave32 per row
static constexpr int TPB  = RPB * 32;

// canonical c4 opcodes
enum {
  OP_LEA = 0, OP_IMM = 1, OP_JMP = 2, OP_JSR = 3, OP_BZ = 4, OP_BNZ = 5,
  OP_ENT = 6, OP_ADJ = 7, OP_LEV = 8, OP_LI = 9, OP_LC = 10, OP_SI = 11,
  OP_SC = 12, OP_PSH = 13,
  OP_OR = 14, OP_XOR = 15, OP_AND = 16, OP_EQ = 17, OP_NE = 18, OP_LT = 19,
  OP_GT = 20, OP_LE = 21, OP_GE = 22, OP_SHL = 23, OP_SHR = 24, OP_ADD = 25,
  OP_SUB = 26, OP_MUL = 27, OP_DIV = 28, OP_MOD = 29
};

// soft read_memory collapses to a bounds-checked gather (eq_gate sums to 0 OOB)
__device__ __forceinline__ float rd_row(const float* __restrict__ row, float addr) {
  const int i = (int)rintf(addr);
  return (i >= 0 && i < MEMN) ? row[i] : 0.0f;
}

__global__ __launch_bounds__(TPB) void c4_step_kernel(
    const float* __restrict__ pc_in, const float* __restrict__ sp_in,
    const float* __restrict__ bp_in, const float* __restrict__ ax_in,
    const float* __restrict__ mem_in, float* __restrict__ out, int B)
{
  const int wave = threadIdx.x >> 5;
  const int lane = threadIdx.x & 31;
  const int b    = blockIdx.x * RPB + wave;
  if (b >= B) return;

  const float* row  = mem_in + (size_t)b * MEMN;
  float*       orow = out + (size_t)4 * B + (size_t)b * MEMN;

#if DEV_ASYNC
  __shared__ float lds[RPB][MEMN];
  // Kick off the bulk 2KB row copy global->LDS with the async DMA path
  // (no VGPR round-trip); it runs while we do the scalar VM step below.
  // 4 issues x 32 lanes x 16B = 2KB, tracked by ASYNCcnt.
  #pragma unroll
  for (int i = 0; i < MEMN / 128; ++i) {
    const float* g = row + i * 128 + lane * 4;
    float*       l = &lds[wave][i * 128 + lane * 4];
    __builtin_amdgcn_global_load_async_to_lds_b128(
        (GAS v4i*)(uintptr_t)g,
        (LAS v4i*)(uint32_t)(uintptr_t)l,   // flat LDS addr: offset in low 32b
        0, 0);
  }
#endif

  // ---------------- discrete VM step (== round(soft step)) ----------------
  // All gates saturate to 0/1 on integer inputs; swiglu_mul(a,b) == a*b.
  // Uniform across the wave (one row per wave).
  const float pc = pc_in[b], sp = sp_in[b], bp = bp_in[b], ax = ax_in[b];

  const float instr = rd_row(row, pc);
  const float imm   = floorf(instr * (1.0f / 256.0f));
  const float opf   = instr - 256.0f * imm;
  const int   op    = (int)rintf(opf);
  const float st    = rd_row(row, sp);
  const bool  pops  = (op >= OP_OR && op <= OP_MOD);

  // ---- new AX (soft MoE -> exact opcode select) ----
  float nax;
  if      (op == OP_ADD) nax = st + ax;
  else if (op == OP_SUB) nax = st - ax;
  else if (op == OP_MUL) nax = st * ax;
  else if (op == OP_DIV || op == OP_MOD) {
    // gate q fires iff st >= q*axs - 0.5 and (q+1)*axs > st + 0.5, q in [0,64)
    const float axs = (ax == 0.0f) ? 1e-8f : ax;
    float q = floorf((st + 0.5f) / axs);
    q = (q >= 0.0f && q <= 63.0f) ? q : 0.0f;
    nax = (op == OP_DIV) ? q : (st - q * axs);
  }
  else if (op == OP_EQ)  nax = (st == ax) ? 1.0f : 0.0f;
  else if (op == OP_NE)  nax = (st != ax) ? 1.0f : 0.0f;
  else if (op == OP_LT)  nax = (st <  ax) ? 1.0f : 0.0f;
  else if (op == OP_GT)  nax = (st >  ax) ? 1.0f : 0.0f;
  else if (op == OP_LE)  nax = (st <= ax) ? 1.0f : 0.0f;
  else if (op == OP_GE)  nax = (st >= ax) ? 1.0f : 0.0f;
  else if (op == OP_SHL || op == OP_SHR) {
    const int k = (int)rintf(ax);
    if (k >= 0 && k < 32) {
      const float p = ldexpf(1.0f, k);
      nax = (op == OP_SHL) ? (st * p) : floorf(st / p);
    } else nax = 0.0f;                 // shift gate fires only for 0<=ax<32
  }
  else if (op == OP_AND) nax = st * ax;                       // soft bitwise on 0/1
  else if (op == OP_OR)  nax = st + ax - st * ax;
  else if (op == OP_XOR) nax = st + ax - 2.0f * st * ax;
  else if (op == OP_IMM) nax = imm;
  else if (op == OP_LEA) nax = bp + imm;
  else if (op == OP_LI || op == OP_LC) nax = rd_row(row, ax);
  else nax = ax;                                              // residual term

  // ---- new SP / BP ----
  float nsp = sp;
  if      (op == OP_PSH) nsp = sp - 8.0f;
  else if (op == OP_ADJ) nsp = sp + imm;
  else if (op == OP_ENT) nsp = sp - imm;
  else if (op == OP_LEV) nsp = bp;
  else if (pops)         nsp = sp + 8.0f;

  float nbp = bp;
  if      (op == OP_ENT) nbp = sp;
  else if (op == OP_LEV) nbp = rd_row(row, bp);

  // ---- new PC ----
  const float pcn = pc + 8.0f;
  float npc = pcn;
  const bool az = (ax == 0.0f);
  if      (op == OP_JMP || op == OP_JSR)  npc = imm;
  else if (op == OP_BZ  &&  az)           npc = imm;
  else if (op == OP_BNZ && !az)           npc = imm;
  else if (op == OP_LEV)                  npc = rd_row(row, sp + 8.0f);

  // ---- memory patches (opcodes mutually exclusive; last-writer-wins order) ----
  const bool wr1 = (op == OP_PSH || op == OP_JSR || op == OP_ENT);
  const int  a1  = (int)rintf(sp - 8.0f);
  const float v1 = (op == OP_PSH) ? ax : ((op == OP_JSR) ? pcn : bp);
  const bool g1  = wr1 && (a1 >= 0) && (a1 < MEMN);

  const bool wr2 = (op == OP_SI || op == OP_SC);
  const int  a2  = (int)rintf(st);
  const float v2 = ax;
  const bool g2  = wr2 && (a2 >= 0) && (a2 < MEMN);

  if (lane == 0) {
    out[b]                 = npc;
    out[(size_t)B + b]     = nsp;
    out[(size_t)2 * B + b] = nbp;
    out[(size_t)3 * B + b] = nax;
  }

#if DEV_ASYNC
  // Row is now (or soon) resident in LDS: drain async loads, patch, store out.
  asm volatile("s_wait_asynccnt 0x0" ::: "memory");
  if (lane == 0) {
    if (g1) lds[wave][a1] = v1;       // ds_store
    if (g2) lds[wave][a2] = v2;       // applied after patch 1 (overrides)
  }
  asm volatile("s_wait_dscnt 0x0" ::: "memory");  // LDS patch visible to async engine
  #pragma unroll
  for (int i = 0; i < MEMN / 128; ++i) {
    float* g = orow + i * 128 + lane * 4;
    float* l = &lds[wave][i * 128 + lane * 4];
    __builtin_amdgcn_global_store_async_from_lds_b128(
        (GAS v4i*)(uintptr_t)g,
        (LAS v4i*)(uint32_t)(uintptr_t)l, 0, 0);
  }
  asm volatile("s_wait_asynccnt 0x0" ::: "memory");
#else
  // Fallback: direct vectorized copy with inline patch merge (b128 ld/st).
  #pragma unroll
  for (int i = 0; i < MEMN / 128; ++i) {
    const int base = i * 128 + lane * 4;
    float4 v = *(const float4*)(row + base);
    float* c = &v.x;
    if (g1 && a1 >= base && a1 < base + 4) c[a1 - base] = v1;
    if (g2 && a2 >= base && a2 < base + 4) c[a2 - base] = v2;
    *(float4*)(orow + base) = v;
  }
#endif
}

extern "C" void kernel_launch(void* const* d_in, const int* in_sizes, int n_in,
                              void* d_out, int out_size, void* d_ws, size_t ws_size,
                              hipStream_t stream) {
  (void)n_in; (void)out_size; (void)d_ws; (void)ws_size;
  const float* pc = (const float*)d_in[0];
  const float* sp = (const float*)d_in[1];
  const float* bp = (const float*)d_in[2];
  const float* ax = (const float*)d_in[3];
  const float* mem = (const float*)d_in[4];
  const int B = in_sizes[0];                  // 32768; in_sizes[4] == B*512
  const int grid = (B + RPB - 1) / RPB;
  c4_step_kernel<<<grid, TPB, 0, stream>>>(pc, sp, bp, ax, mem, (float*)d_out, B);
}